// GATLayer_72060961292568
// MI455X (gfx1250) — compile-verified
//
#include <hip/hip_runtime.h>
#include <math.h>

// ---------------------------------------------------------------------------
// GAT layer for MI455X (gfx1250, wave32, WMMA).
//   h = x @ W (fp32 WMMA 16x16x4), per-head attention dots, 3-pass
//   scatter-softmax over edges (+self loops), weighted scatter aggregation,
//   LayerNorm + ELU.
// ---------------------------------------------------------------------------

#define N_NODES 50000
#define N_EDGES 1600000
#define ET_EDGES (N_EDGES + N_NODES)   // edges + self loops
#define IN_CH 128
#define OUT_CH 128
#define HEADS 4
#define CPH 32                          // channels per head
#define NEG_SLOPE 0.2f
#define SOFTMAX_EPS 1e-16f
#define LN_EPS 1e-5f

typedef float v2f __attribute__((ext_vector_type(2)));
typedef float v8f __attribute__((ext_vector_type(8)));

// ---------------------------------------------------------------------------
// Kernel 0: zero-init output accumulator, segment sums, segment-max keys
// ---------------------------------------------------------------------------
__global__ void k_init(float* __restrict__ out, float* __restrict__ ssum,
                       unsigned* __restrict__ mkey, int nout, int nseg) {
    int i = blockIdx.x * blockDim.x + threadIdx.x;
    if (i < nout) out[i] = 0.0f;
    if (i < nseg) { ssum[i] = 0.0f; mkey[i] = 0u; }   // 0u == ordered -inf
}

// ---------------------------------------------------------------------------
// Kernel 1: h = x @ W  via V_WMMA_F32_16X16X4_F32.
// Block = 256 threads (8 waves). Each block: 128 rows; each wave: 16 rows x
// 128 cols (8 C-tiles of 16x16 f32 = 8 v8f accumulators).
// W (128x128 f32 = 64 KB) is staged in LDS once per block.
// A frag (16x4 f32): lane l<16 -> M=l, K={0,1}; lane>=16 -> M=l-16, K={2,3}.
// B frag (4x16 f32): vgpr0 lanes0-15 K=0 / lanes16-31 K=2; vgpr1 K=1/K=3.
// Inner loop: A(k+4) software-pipelined; all 8 B frags fetched as single
// ds_load_2addr_b32 each (offsets t*16 and 128+t*16 dwords -> pair lands in
// adjacent VGPRs = the WMMA operand, no shuffles), then an 8-WMMA burst on
// independent accumulators (no D->A/B hazards).
// ---------------------------------------------------------------------------
__global__ void k_gemm_wmma(const float* __restrict__ x,
                            const float* __restrict__ W,
                            float* __restrict__ h, int nrows) {
    extern __shared__ float Wlds[];   // 128*128 floats = 64 KB
    for (int i = threadIdx.x; i < (IN_CH * OUT_CH) / 4; i += blockDim.x)
        ((float4*)Wlds)[i] = ((const float4*)W)[i];
    __syncthreads();

    const int wave = threadIdx.x >> 5;
    const int lane = threadIdx.x & 31;
    const int rowBase = blockIdx.x * 128 + wave * 16;
    const int m  = lane & 15;
    const int kk = (lane >> 4) << 1;        // 0 or 2

    int arow = rowBase + m;
    if (arow >= nrows) arow = nrows - 1;    // clamp (stores are guarded)
    const float* xk = x + (size_t)arow * IN_CH + kk;

    v8f c[8];
#pragma unroll
    for (int t = 0; t < 8; ++t) c[t] = (v8f){0,0,0,0,0,0,0,0};

    float2 av = *(const float2*)xk;                 // A prefetch for k = 0
    for (int k = 0; k < IN_CH; k += 4) {
        v2f a; a.x = av.x; a.y = av.y;
        if (k + 4 < IN_CH) av = *(const float2*)(xk + k + 4);   // pipeline A

        const float* w0 = &Wlds[(k + kk) * OUT_CH + m];
        v2f b[8];
#pragma unroll
        for (int t = 0; t < 8; ++t) {
            b[t].x = w0[t * 16];                    // row k+kk
            b[t].y = w0[OUT_CH + t * 16];           // row k+kk+1
        }
#pragma unroll
        for (int t = 0; t < 8; ++t)
            c[t] = __builtin_amdgcn_wmma_f32_16x16x4_f32(
                false, a, false, b[t], (short)0, c[t], false, false);
    }

#pragma unroll
    for (int t = 0; t < 8; ++t) {
#pragma unroll
        for (int v = 0; v < 8; ++v) {
            const int M = v + ((lane >> 4) << 3);
            const int row = rowBase + M;
            if (row < nrows)
                h[(size_t)row * OUT_CH + t * 16 + m] = c[t][v];
        }
    }
}

// ---------------------------------------------------------------------------
// Kernel 2: a_s[n,h] = h[n,h,:] . att_src[h,:]  (and a_d with att_dst).
// Block = 256 threads = 8 waves -> 2 nodes/block, wave = one (node,head).
// ---------------------------------------------------------------------------
__global__ void k_dots(const float* __restrict__ h,
                       const float* __restrict__ att_src,
                       const float* __restrict__ att_dst,
                       float* __restrict__ a_s, float* __restrict__ a_d, int n) {
    const int wave = threadIdx.x >> 5;
    const int lane = threadIdx.x & 31;
    const int node = blockIdx.x * 2 + (wave >> 2);
    const int head = wave & 3;
    if (node >= n) return;
    const float hv = h[(size_t)node * OUT_CH + head * CPH + lane];
    float vs = hv * att_src[head * CPH + lane];
    float vd = hv * att_dst[head * CPH + lane];
#pragma unroll
    for (int o = 16; o > 0; o >>= 1) {
        vs += __shfl_xor(vs, o, 32);
        vd += __shfl_xor(vd, o, 32);
    }
    if (lane == 0) {
        a_s[node * HEADS + head] = vs;
        a_d[node * HEADS + head] = vd;
    }
}

// ---------------------------------------------------------------------------
// Kernel 3: per (edge,head) leaky-relu logit + segment max (order-preserving
// uint atomicMax). Edges e >= E are self loops (src = dst = e - E).
// ---------------------------------------------------------------------------
__device__ __forceinline__ unsigned f2key(float f) {
    unsigned b = __float_as_uint(f);
    return (b & 0x80000000u) ? ~b : (b | 0x80000000u);
}
__device__ __forceinline__ float key2f(unsigned k) {
    unsigned b = (k & 0x80000000u) ? (k ^ 0x80000000u) : ~k;
    return __uint_as_float(b);
}

__global__ void k_edge_max(const int* __restrict__ ei,
                           const float* __restrict__ a_s,
                           const float* __restrict__ a_d,
                           float* __restrict__ logits,
                           unsigned* __restrict__ mkey) {
    const int tid = blockIdx.x * blockDim.x + threadIdx.x;
    if (tid >= ET_EDGES * HEADS) return;
    const int e  = tid >> 2;
    const int hd = tid & 3;
    int s, d;
    if (e < N_EDGES) { s = ei[e]; d = ei[N_EDGES + e]; }
    else             { s = d = e - N_EDGES; }
    float v = a_s[s * HEADS + hd] + a_d[d * HEADS + hd];
    v = v > 0.0f ? v : NEG_SLOPE * v;
    logits[tid] = v;
    atomicMax(&mkey[d * HEADS + hd], f2key(v));
}

// ---------------------------------------------------------------------------
// Kernel 4: e = exp(logit - m[dst]); segment sum via atomicAdd.
// ---------------------------------------------------------------------------
__global__ void k_edge_exp(const int* __restrict__ ei,
                           float* __restrict__ logits,
                           const unsigned* __restrict__ mkey,
                           float* __restrict__ ssum) {
    const int tid = blockIdx.x * blockDim.x + threadIdx.x;
    if (tid >= ET_EDGES * HEADS) return;
    const int e  = tid >> 2;
    const int hd = tid & 3;
    const int d = (e < N_EDGES) ? ei[N_EDGES + e] : (e - N_EDGES);
    const float mf = key2f(mkey[d * HEADS + hd]);
    const float ex = expf(logits[tid] - mf);
    logits[tid] = ex;
    atomicAdd(&ssum[d * HEADS + hd], ex);
}

// ---------------------------------------------------------------------------
// Kernel 5: wave32 per edge: out[dst] += h[src] * alpha. Each lane carries
// 4 contiguous channels (float4 -> fully coalesced 512B gather per wave).
// ---------------------------------------------------------------------------
__global__ void k_edge_agg(const int* __restrict__ ei,
                           const float* __restrict__ h,
                           const float* __restrict__ evals,
                           const float* __restrict__ ssum,
                           float* __restrict__ out) {
    const int wave = threadIdx.x >> 5;
    const int lane = threadIdx.x & 31;
    const int e = blockIdx.x * 8 + wave;
    if (e >= ET_EDGES) return;
    int s, d;
    if (e < N_EDGES) { s = ei[e]; d = ei[N_EDGES + e]; }
    else             { s = d = e - N_EDGES; }
    const int head = lane >> 3;                      // (lane*4)/32
    const float alpha = evals[e * HEADS + head] /
                        (ssum[d * HEADS + head] + SOFTMAX_EPS);
    const float4 hv = *(const float4*)(h + (size_t)s * OUT_CH + lane * 4);
    float* op = out + (size_t)d * OUT_CH + lane * 4;
    atomicAdd(op + 0, hv.x * alpha);
    atomicAdd(op + 1, hv.y * alpha);
    atomicAdd(op + 2, hv.z * alpha);
    atomicAdd(op + 3, hv.w * alpha);
}

// ---------------------------------------------------------------------------
// Kernel 6: +bias, LayerNorm over 128 channels, gamma/beta, ELU. In place.
// Block = 128 threads (4 waves), wave32 shuffle + tiny LDS cross-wave reduce.
// ---------------------------------------------------------------------------
__global__ void k_ln_elu(float* __restrict__ out,
                         const float* __restrict__ bias,
                         const float* __restrict__ gamma,
                         const float* __restrict__ beta) {
    __shared__ float red[4];
    const int node = blockIdx.x;
    const int c = threadIdx.x;
    const int wave = c >> 5, lane = c & 31;

    float v = out[(size_t)node * OUT_CH + c] + bias[c];

    float t = v;
#pragma unroll
    for (int o = 16; o > 0; o >>= 1) t += __shfl_xor(t, o, 32);
    if (lane == 0) red[wave] = t;
    __syncthreads();
    const float mu = (red[0] + red[1] + red[2] + red[3]) * (1.0f / OUT_CH);
    __syncthreads();

    const float dv = v - mu;
    t = dv * dv;
#pragma unroll
    for (int o = 16; o > 0; o >>= 1) t += __shfl_xor(t, o, 32);
    if (lane == 0) red[wave] = t;
    __syncthreads();
    const float var = (red[0] + red[1] + red[2] + red[3]) * (1.0f / OUT_CH);

    float y = dv * (1.0f / sqrtf(var + LN_EPS)) * gamma[c] + beta[c];
    y = y > 0.0f ? y : (expf(y) - 1.0f);
    out[(size_t)node * OUT_CH + c] = y;
}

// ---------------------------------------------------------------------------
// Host launcher
// ---------------------------------------------------------------------------
extern "C" void kernel_launch(void* const* d_in, const int* in_sizes, int n_in,
                              void* d_out, int out_size, void* d_ws, size_t ws_size,
                              hipStream_t stream) {
    const float* x       = (const float*)d_in[0];
    const int*   ei      = (const int*)  d_in[1];
    const float* W       = (const float*)d_in[2];
    const float* att_src = (const float*)d_in[3];
    const float* att_dst = (const float*)d_in[4];
    const float* bias    = (const float*)d_in[5];
    const float* gamma   = (const float*)d_in[6];
    const float* beta    = (const float*)d_in[7];
    float* out = (float*)d_out;

    // Workspace layout (all sizes multiples of 256 B)
    char* ws = (char*)d_ws;
    float*    h      = (float*)ws;                               // 25.6 MB
    ws += (size_t)N_NODES * OUT_CH * sizeof(float);
    float*    a_s    = (float*)ws;  ws += (size_t)N_NODES * HEADS * sizeof(float);
    float*    a_d    = (float*)ws;  ws += (size_t)N_NODES * HEADS * sizeof(float);
    unsigned* mkey   = (unsigned*)ws; ws += (size_t)N_NODES * HEADS * sizeof(unsigned);
    float*    ssum   = (float*)ws;  ws += (size_t)N_NODES * HEADS * sizeof(float);
    float*    evals  = (float*)ws;  // logits -> exp values, 26.4 MB

    const int nout = N_NODES * OUT_CH;         // 6.4M
    const int nseg = N_NODES * HEADS;          // 200K

    // 0) init accumulators
    k_init<<<(nout + 255) / 256, 256, 0, stream>>>(out, ssum, mkey, nout, nseg);

    // 1) h = x @ W (WMMA f32)
    k_gemm_wmma<<<(N_NODES + 127) / 128, 256, IN_CH * OUT_CH * sizeof(float),
                  stream>>>(x, W, h, N_NODES);

    // 2) attention dot products
    k_dots<<<(N_NODES + 1) / 2, 256, 0, stream>>>(h, att_src, att_dst, a_s, a_d,
                                                  N_NODES);

    // 3) logits + segment max
    k_edge_max<<<(ET_EDGES * HEADS + 255) / 256, 256, 0, stream>>>(
        ei, a_s, a_d, evals, mkey);

    // 4) exp + segment sum
    k_edge_exp<<<(ET_EDGES * HEADS + 255) / 256, 256, 0, stream>>>(
        ei, evals, mkey, ssum);

    // 5) weighted scatter aggregation (wave per edge)
    k_edge_agg<<<(ET_EDGES + 7) / 8, 256, 0, stream>>>(ei, h, evals, ssum, out);

    // 6) bias + LayerNorm + ELU (in place on d_out)
    k_ln_elu<<<N_NODES, 128, 0, stream>>>(out, bias, gamma, beta);
}